// MultilevelROIAligner_44435731644919
// MI455X (gfx1250) — compile-verified
//
#include <hip/hip_runtime.h>
#include <hip/hip_bf16.h>

#define CROP 7
#define NBOX 256
#define CHAN 256
#define CT   64   // channel tile held in LDS per pass

typedef __attribute__((ext_vector_type(2))) float v2f;
typedef __attribute__((ext_vector_type(4))) float v4f;
typedef __attribute__((ext_vector_type(8))) float v8f;

// One workgroup per ROI (B*N = 2048 blocks, 256 threads = 8 wave32).
// Phase A: gather + x-interp with b128 global loads (bandwidth-bound part).
// Phase B: y-interp as exact fp32 matmul on v_wmma_f32_16x16x4_f32:
//   Out[p, c] = sum_y KY[p, y] * U[y, q, c],  KY 7x14 zero-padded to 16x16.
__global__ __launch_bounds__(256)
void MultilevelROIAligner_wmma_kernel(const float* __restrict__ f2,
                                      const float* __restrict__ f3,
                                      const float* __restrict__ f4,
                                      const float* __restrict__ f5,
                                      const float* __restrict__ boxes,
                                      float* __restrict__ out)
{
    __shared__ float s_ky_pad[16];        // flat [p][i] = ky weight, rows 14/15 = 0
    __shared__ float s_kx[CROP][2];
    __shared__ int   s_yi[CROP][2];
    __shared__ int   s_xi[CROP][2];
    __shared__ int   s_lev0;
    __shared__ float s_U[16][CROP][CT];   // K-rows padded to 16; rows 14/15 stay 0

    const int bn  = blockIdx.x;           // b*NBOX + n
    const int b   = bn / NBOX;
    const int tid = threadIdx.x;

    // ---- Phase 0: box -> level, grid positions, bilinear weights ----
    if (tid == 0) {
        const float* bx = boxes + (size_t)bn * 4;
        float y1 = bx[0], x1 = bx[1], y2 = bx[2], x2 = bx[3];
        float bh = y2 - y1, bw = x2 - x1;
        float areas_sqrt = sqrtf(bh * bw);
        float levf = floorf(log2f(areas_sqrt / 224.0f)) + 4.0f;
        levf = fminf(fmaxf(levf, 2.0f), 5.0f);
        int   lev   = (int)levf;
        float scale = (float)(1 << lev);
        float y0b = y1 / scale, x0b = x1 / scale;
        bh /= scale; bw /= scale;
        int   lev0 = lev - 2;
        float bnd  = 256.0f / (float)(1 << lev0) - 1.0f;  // H0 == W0 == 256
        for (int s = 0; s < CROP; ++s) {
            float gy  = y0b + (s + 0.5f) * bh / 7.0f;
            float fy0 = fminf(fmaxf(floorf(gy), 0.0f), bnd);
            float fy1 = fminf(fy0 + 1.0f, bnd);
            float ly  = gy - fy0;
            s_ky_pad[2 * s]     = 1.0f - ly;
            s_ky_pad[2 * s + 1] = ly;
            s_yi[s][0] = (int)fy0;  s_yi[s][1] = (int)fy1;

            float gx  = x0b + (s + 0.5f) * bw / 7.0f;
            float fx0 = fminf(fmaxf(floorf(gx), 0.0f), bnd);
            float fx1 = fminf(fx0 + 1.0f, bnd);
            float lx  = gx - fx0;
            s_kx[s][0] = 1.0f - lx; s_kx[s][1] = lx;
            s_xi[s][0] = (int)fx0;  s_xi[s][1] = (int)fx1;
        }
        s_ky_pad[14] = 0.0f;
        s_ky_pad[15] = 0.0f;
        s_lev0 = lev0;
    }
    // Zero the K-padding rows of U once; Phase A never touches rows 14/15.
    {
        float* up = &s_U[14][0][0];           // 2 * 7 * CT = 896 floats
        for (int i = tid; i < 2 * CROP * CT; i += 256) up[i] = 0.0f;
    }
    __syncthreads();

    const int lev0 = s_lev0;
    const float* feat; int W;
    switch (lev0) {
        case 0:  feat = f2; W = 256; break;
        case 1:  feat = f3; W = 128; break;
        case 2:  feat = f4; W =  64; break;
        default: feat = f5; W =  32; break;
    }
    const size_t batch_off = (size_t)b * W * W * CHAN;

    const int lane = tid & 31;
    const int wave = tid >> 5;
    const int half = lane >> 4;     // 0: lanes 0-15 (K=v), 1: lanes 16-31 (K=v+2)
    const int ln   = lane & 15;
    const size_t out_base = (size_t)bn * (CROP * CROP * CHAN);

    // ---- Precompute loop-invariant A fragments (KY, 16x16, 4 k-steps) ----
    // A[16x4] fragment layout: M = ln, K = kk*4 + half*2 + {0,1}.
    // KY[p, y] = s_ky_pad[y] iff (y >> 1) == p, else 0. Unconditional LDS
    // read from the padded array + select (no EXEC-divergent branches).
    v2f a_frag[4];
    #pragma unroll
    for (int kk = 0; kk < 4; ++kk) {
        const int k0 = kk * 4 + half * 2;
        float t0 = s_ky_pad[k0];
        float t1 = s_ky_pad[k0 + 1];
        a_frag[kk].x = (( k0      >> 1) == ln) ? t0 : 0.0f;
        a_frag[kk].y = (((k0 + 1) >> 1) == ln) ? t1 : 0.0f;
    }

    for (int ct = 0; ct < CHAN / CT; ++ct) {
        const int cbase = ct * CT;

        // ---- Phase A: gather + x-interp (vectorized float4 along channels)
        //   U[y][q][c] = kx0 * F(row_y, x0_q, c) + kx1 * F(row_y, x1_q, c)
        {
            const int c4 = (tid & 15) * 4;            // 16 threads cover CT=64
            for (int j = tid >> 4; j < 14 * CROP; j += 16) {
                const int y   = j / CROP;
                const int q   = j - y * CROP;
                const int row = s_yi[y >> 1][y & 1];
                const size_t rbase = batch_off + (size_t)row * W * CHAN + cbase + c4;
                const v4f v0 = *(const v4f*)&feat[rbase + (size_t)s_xi[q][0] * CHAN];
                const v4f v1 = *(const v4f*)&feat[rbase + (size_t)s_xi[q][1] * CHAN];
                const float w0 = s_kx[q][0], w1 = s_kx[q][1];
                v4f u;
                u.x = w0 * v0.x + w1 * v1.x;
                u.y = w0 * v0.y + w1 * v1.y;
                u.z = w0 * v0.z + w1 * v1.z;
                u.w = w0 * v0.w + w1 * v1.w;
                *(v4f*)&s_U[y][q][c4] = u;
            }
        }
        __syncthreads();

        // ---- Phase B: y-interp via WMMA f32 16x16x4 (exact fp32) ----
        // 7 q positions x 4 channel-16 sub-tiles = 28 jobs across 8 waves.
        // Wave-uniform trip counts -> EXEC all-ones at every WMMA.
        for (int job = wave; job < CROP * (CT / 16); job += 8) {
            const int q   = job >> 2;
            const int c16 = job & 3;
            v8f acc = {};
            #pragma unroll
            for (int kk = 0; kk < 4; ++kk) {
                const int k0 = kk * 4 + half * 2;
                // B[4x16] fragment: K-rows from padded U, N = ln (channel).
                v2f bf;
                bf.x = s_U[k0    ][q][c16 * 16 + ln];
                bf.y = s_U[k0 + 1][q][c16 * 16 + ln];
                acc = __builtin_amdgcn_wmma_f32_16x16x4_f32(
                          /*neg_a=*/false, a_frag[kk], /*neg_b=*/false, bf,
                          /*c_mod=*/(short)0, acc,
                          /*reuse_a=*/false, /*reuse_b=*/false);
            }
            // D layout: vgpr v -> M=v (lanes 0-15) / M=v+8 (lanes 16-31).
            // Only rows p = 0..6 are real output.
            if (half == 0) {
                #pragma unroll
                for (int p = 0; p < CROP; ++p) {
                    out[out_base + (size_t)(p * CROP + q) * CHAN + cbase + c16 * 16 + ln]
                        = acc[p];
                }
            }
        }
        __syncthreads();
    }
}

extern "C" void kernel_launch(void* const* d_in, const int* in_sizes, int n_in,
                              void* d_out, int out_size, void* d_ws, size_t ws_size,
                              hipStream_t stream) {
    const float* f2    = (const float*)d_in[0];
    const float* f3    = (const float*)d_in[1];
    const float* f4    = (const float*)d_in[2];
    const float* f5    = (const float*)d_in[3];
    const float* boxes = (const float*)d_in[4];
    float* out = (float*)d_out;

    const int nroi = in_sizes[4] / 4;   // B*N = 2048
    MultilevelROIAligner_wmma_kernel<<<nroi, 256, 0, stream>>>(
        f2, f3, f4, f5, boxes, out);
    (void)d_ws; (void)ws_size; (void)n_in; (void)out_size;
}